// TokenEmbeddingWithFeatures_7413113553148
// MI455X (gfx1250) — compile-verified
//
#include <hip/hip_runtime.h>
#include <cstdint>
#include <cstddef>

// ---------------------------------------------------------------------------
// Token embedding + positional encoding, MI455X (gfx1250).
//
// Store-bound kernel: 64*2048*768 f32 = ~403 MB out -> ~17us floor @23.3TB/s.
// CDNA5 features used:
//   * global_load_async_to_lds_b128 (ASYNCcnt) staging the ~31KB of embedding
//     tables into LDS, overlapped with the per-block sincos PE computation.
//   * s_wait_asynccnt to fence the async fill.
//   * non-temporal b128 output stores (write-once 403MB stream > 192MB L2).
// No matmul exists in this op, so WMMA is intentionally not used.
// ---------------------------------------------------------------------------

#define S_LEN    2048
#define BATCH    64
#define DMODEL   768
#define NTHREADS 192   // 6 waves; thread t owns dims 4t..4t+3

typedef float v4f __attribute__((ext_vector_type(4)));  // native clang vector

__global__ __launch_bounds__(NTHREADS)
void token_embed_pe_kernel(const int*   __restrict__ tokens,     // [64, 2048]
                           const float* __restrict__ color_W,    // [4, 256]
                           const float* __restrict__ shape_W,    // [4, 256]
                           const float* __restrict__ quantity_W, // [4, 256]
                           const float* __restrict__ special_W,  // [6, 768]
                           float*       __restrict__ out)        // [64,2048,768]
{
  // Single contiguous LDS block: [color|shape|quantity] (3*1024 f) then
  // special (6*768 f).  Byte offsets: feat @ 0, spec @ 12288.
  __shared__ __align__(16) float smem[3 * 1024 + 6 * DMODEL]; // 30720 B
  __shared__ int tok_s[BATCH];

  const int tid = threadIdx.x;
  const int s   = blockIdx.x;

  float* const feat = smem;            // 3 regions of 4x256
  float* const spec = smem + 3 * 1024; // 6x768

  // Low 32 bits of a flat LDS pointer == LDS byte offset (ISA 10.2 aperture).
  const uint32_t smem_off = (uint32_t)(uintptr_t)(void*)smem;

  // --- Phase A: async table fill, flat chunk space (ASYNCcnt) --------------
  // chunks (float4): [0,256) color | [256,512) shape | [512,768) quantity |
  //                  [768,1920) special.   1920 chunks / 192 thr = 10 each.
  #pragma unroll
  for (int i = 0; i < 10; ++i) {
    const int c = tid + i * NTHREADS;          // 0..1919, no bounds check
    const float* src;
    uint32_t gbase;
    if (c < 256)       { src = color_W;    gbase = (uint32_t)c * 16u; }
    else if (c < 512)  { src = shape_W;    gbase = (uint32_t)(c - 256) * 16u; }
    else if (c < 768)  { src = quantity_W; gbase = (uint32_t)(c - 512) * 16u; }
    else               { src = special_W;  gbase = (uint32_t)(c - 768) * 16u; }
    const uint32_t dst = smem_off + (uint32_t)c * 16u;  // LDS layout == chunk order
    asm volatile("global_load_async_to_lds_b128 %0, %1, %2"
                 :: "v"(dst), "v"(gbase), "s"(src)
                 : "memory");
  }

  // Stage this position's 64 tokens (one per batch row) into LDS.
  if (tid < BATCH) tok_s[tid] = tokens[(size_t)tid * S_LEN + s];

  // --- Phase B: positional encoding for dims d0..d0+3 (overlaps async) -----
  // pe[s,2k] = sin(s * 10000^(-2k/768)), pe[s,2k+1] = cos(same angle).
  // Thread t covers pairs k = 2t (dims 4t,4t+1) and k = 2t+1 (dims 4t+2,4t+3).
  const int   d0 = tid * 4;
  const float L  = 9.21034037197618f; // ln(10000)
  const float inv0 = __expf(-((float)(d0)     * (1.0f / 768.f)) * L);
  const float inv1 = __expf(-((float)(d0 + 2) * (1.0f / 768.f)) * L);
  float s0, c0, s1, c1;
  __sincosf((float)s * inv0, &s0, &c0);
  __sincosf((float)s * inv1, &s1, &c1);
  v4f pe;  pe.x = s0; pe.y = c0; pe.z = s1; pe.w = c1;

  // Feature-region selection: wave-uniform (boundaries at d=256,512 align
  // with wave boundaries for 192 threads).
  const int region = d0 >> 8;                                  // 0,1,2
  const int shift  = (region == 0) ? 4 : (region == 1 ? 2 : 0);// color/shape/qty
  const float* const featR = feat + region * 1024 + (d0 & 255);
  const float* const specR = spec + d0;

  // --- Fence async fill, then sync all waves ------------------------------
  asm volatile("s_wait_asynccnt 0x0" ::: "memory");
  __syncthreads();

  const float SCALE = 27.712812921102035f; // sqrt(768)

  // --- Phase C: 64 coalesced 3KB row writes (store-bound inner loop) -------
  float* outp = out + (size_t)s * DMODEL + d0;   // advance by 6MB per b
  #pragma unroll 4
  for (int b = 0; b < BATCH; ++b) {
    const int tok = tok_s[b];             // uniform across the block
    v4f e;
    if (tok < 64) {
      const int sub = (tok >> shift) & 3; // color=t>>4, shape=(t>>2)&3, qty=t&3
      e = *(const v4f*)(featR + sub * 256);
    } else {
      int sp = tok - 64;
      if (sp > 5) sp = 5;
      e = *(const v4f*)(specR + sp * DMODEL);
    }
    const v4f r = e * SCALE + pe;
    // Write-once stream: keep it out of L2 (th:NT).
    __builtin_nontemporal_store(r, (v4f*)outp);
    outp += (size_t)S_LEN * DMODEL;
  }
}

extern "C" void kernel_launch(void* const* d_in, const int* in_sizes, int n_in,
                              void* d_out, int out_size, void* d_ws, size_t ws_size,
                              hipStream_t stream) {
  (void)in_sizes; (void)n_in; (void)out_size; (void)d_ws; (void)ws_size;

  const int*   tokens     = (const int*)  d_in[0]; // [64,2048] (integer -> int32)
  const float* color_W    = (const float*)d_in[1]; // [4,256]
  const float* shape_W    = (const float*)d_in[2]; // [4,256]
  const float* quantity_W = (const float*)d_in[3]; // [4,256]
  const float* special_W  = (const float*)d_in[4]; // [6,768]
  float*       out        = (float*)d_out;         // [64,2048,768]

  token_embed_pe_kernel<<<dim3(S_LEN), dim3(NTHREADS), 0, stream>>>(
      tokens, color_W, shape_W, quantity_W, special_W, out);
}